// FilteredInstanceNorm1D_30829275251391
// MI455X (gfx1250) — compile-verified
//
#include <hip/hip_runtime.h>

typedef __attribute__((ext_vector_type(2))) float v2f;
typedef __attribute__((ext_vector_type(4))) float v4f;
typedef __attribute__((ext_vector_type(8))) float v8f;

#define FIN_EPS 1e-5f

constexpr int T_DIM     = 4096;
constexpr int CH_PER_WG = 16;
constexpr int THREADS   = 256;
constexpr int WAVES     = THREADS / 32;      // 8 waves (wave32)
constexpr int CHUNK     = T_DIM / WAVES;     // 512 t-values per wave
constexpr int ITERS     = CHUNK / 4;         // 128 WMMA steps (K=4 each)

__device__ __forceinline__ v8f wmma_acc(v2f a, v2f b, v8f c) {
    return __builtin_amdgcn_wmma_f32_16x16x4_f32(
        false, a, false, b, (short)0, c, false, false);
}

__global__ __launch_bounds__(THREADS)
void finorm_wmma_kernel(const float* __restrict__ x,
                        const int*   __restrict__ x_lens,
                        float*       __restrict__ out,
                        int C)
{
    const int groups_per_b = C / CH_PER_WG;
    const int b  = blockIdx.x / groups_per_b;
    const int cg = blockIdx.x % groups_per_b;
    const int c0 = cg * CH_PER_WG;
    const int len = x_lens[b];

    const int tid    = threadIdx.x;
    const int wave   = tid >> 5;
    const int lane   = tid & 31;
    const int hi     = lane >> 4;    // 0: K={0,1}, 1: K={2,3}
    const int lane16 = lane & 15;    // channel-within-group (B-matrix column)

    __shared__ float s_sum[WAVES][CH_PER_WG];
    __shared__ float s_sq [WAVES][CH_PER_WG];
    __shared__ float s_mean[CH_PER_WG];
    __shared__ float s_rstd[CH_PER_WG];

    // ------------- Phase 1: masked sum / sumsq via V_WMMA_F32_16X16X4_F32 -------------
    // D[m][n] = sum_k A[m][k]*B[k][n];  A = mask row (1.0 for full steps),
    // B[k][n] = x[c0+n, t+k].  D row 0 col n = masked partial sum for channel c0+n.
    const float* rowbase = x + ((size_t)b * C + (size_t)(c0 + lane16)) * T_DIM;
    const int t0 = wave * CHUNK;

    // Wave-uniform trip counts, forced into SGPRs so loop control is scalar and
    // EXEC stays all-ones around every WMMA (ISA requirement).
    const int rel_s   = __builtin_amdgcn_readfirstlane(len - t0);
    const int nfull_s = rel_s >= CHUNK ? ITERS : (rel_s > 0 ? (rel_s >> 2) : 0);

    v8f acc_s0 = {}; v8f acc_s1 = {};
    v8f acc_q0 = {}; v8f acc_q1 = {};
    const v2f ones = {1.0f, 1.0f};

    const float* p = rowbase + t0 + 2 * hi;

    // Full (unmasked) steps: no compares, constant A = 1.0, two independent chains.
    int i = 0;
    for (; i + 2 <= nfull_s; i += 2) {
        v2f b0 = *(const v2f*)(p);
        v2f b1 = *(const v2f*)(p + 4);
        v2f q0 = b0 * b0;
        v2f q1 = b1 * b1;
        acc_s0 = wmma_acc(ones, b0, acc_s0);
        acc_q0 = wmma_acc(ones, q0, acc_q0);
        acc_s1 = wmma_acc(ones, b1, acc_s1);
        acc_q1 = wmma_acc(ones, q1, acc_q1);
        p += 8;
    }
    if (i < nfull_s) {
        v2f b0 = *(const v2f*)(p);
        acc_s0 = wmma_acc(ones, b0, acc_s0);
        acc_q0 = wmma_acc(ones, b0 * b0, acc_q0);
        ++i;
    }
    // Single boundary step (only wave containing t == len, with len % 4 != 0).
    if (rel_s > 0 && rel_s < CHUNK && (rel_s & 3)) {
        const int k = t0 + 4 * nfull_s + 2 * hi;
        v2f bv = *(const v2f*)(rowbase + k);
        v2f av;
        av.x = (k     < len) ? 1.0f : 0.0f;
        av.y = (k + 1 < len) ? 1.0f : 0.0f;
        acc_s0 = wmma_acc(av, bv, acc_s0);
        acc_q0 = wmma_acc(av, bv * bv, acc_q0);
    }
    // Steps fully beyond len contribute exactly 0: skipped (no loads, no WMMA).

    v8f accs = acc_s0 + acc_s1;
    v8f accq = acc_q0 + acc_q1;

    // D VGPR0: lanes 0-15 hold (M=0, N=lane).  All D rows are identical.
    if (hi == 0) {
        s_sum[wave][lane16] = accs.s0;
        s_sq [wave][lane16] = accq.s0;
    }
    __syncthreads();

    // ------------- Phase 2: combine wave partials -> mean / rstd -------------
    if (tid < CH_PER_WG) {
        float ts = 0.0f, tq = 0.0f;
        #pragma unroll
        for (int w = 0; w < WAVES; ++w) {
            ts += s_sum[w][tid];
            tq += s_sq [w][tid];
        }
        const float n    = (float)len;              // len clipped to [2, T] upstream
        const float mean = ts / n;
        float var = (tq - ts * mean) / (n - 1.0f);  // masked E[(x-mean)^2]*n/(n-1)
        var = var > 0.0f ? var : 0.0f;
        s_mean[tid] = mean;
        s_rstd[tid] = __frsqrt_rn(var + FIN_EPS);
    }
    __syncthreads();

    // ------------- Phase 3: normalize full rows (reference normalizes ALL t) -------------
    {
        const int row = tid >> 4;                   // 0..15 channel in group
        const int col = tid & 15;
        const float mean = s_mean[row];
        const float rstd = s_rstd[row];
        const size_t base = ((size_t)b * C + (size_t)(c0 + row)) * T_DIM;
        const float* rp = x   + base;
        float*       op = out + base;
        #pragma unroll 4
        for (int j = 0; j < T_DIM / (16 * 4); ++j) { // 64 float4 per thread
            const int e = (col + j * 16) * 4;        // 16 threads x 16B = 256B contiguous
            v4f v = __builtin_nontemporal_load((const v4f*)(rp + e)); // last use of x
            v4f o = (v - mean) * rstd;
            __builtin_nontemporal_store(o, (v4f*)(op + e));           // streaming out
        }
    }
}

extern "C" void kernel_launch(void* const* d_in, const int* in_sizes, int n_in,
                              void* d_out, int out_size, void* d_ws, size_t ws_size,
                              hipStream_t stream)
{
    const float* x      = (const float*)d_in[0];
    const int*   x_lens = (const int*)d_in[1];
    float*       out    = (float*)d_out;

    const int B = in_sizes[1];                       // x_lens has B elements
    const int C = (int)((long long)in_sizes[0] / ((long long)B * T_DIM));

    const int grid = B * (C / CH_PER_WG);            // 32 * 16 = 512 workgroups
    finorm_wmma_kernel<<<grid, THREADS, 0, stream>>>(x, x_lens, out, C);
}